// ElementwiseTensorProduct_60696477827399
// MI455X (gfx1250) — compile-verified
//
#include <hip/hip_runtime.h>

typedef __attribute__((ext_vector_type(2))) float v2f;
typedef __attribute__((ext_vector_type(4))) float v4f;
typedef __attribute__((ext_vector_type(8))) float v8f;

#define DIM_IN  480
#define DIM_OUT 1504
#define W1_OFF  0        // [12][16] padded B for l=1 (K rows x out cols)
#define W2_OFF  192      // [28][32] padded B for l=2
#define W_TOTAL (192 + 28*32)

// ===================== device-side e3nn real Wigner-3j =====================

__device__ double factd(int n) { double r = 1.0; for (int i = 2; i <= n; ++i) r *= (double)i; return r; }

__device__ double su2_cg_coeff(int j1, int m1, int j2, int m2, int j3, int m3) {
  if (m3 != m1 + m2) return 0.0;
  int vmin = -j1 + j2 + m3; if (-j1 + m1 > vmin) vmin = -j1 + m1; if (vmin < 0) vmin = 0;
  int vmax = j2 + j3 + m1; if (j3 - j1 + j2 < vmax) vmax = j3 - j1 + j2; if (j3 + m3 < vmax) vmax = j3 + m3;
  double C = sqrt((2.0 * j3 + 1.0)
      * factd(j3 + j1 - j2) * factd(j3 - j1 + j2) * factd(j1 + j2 - j3) * factd(j3 + m3) * factd(j3 - m3)
      / (factd(j1 + j2 + j3 + 1) * factd(j1 - m1) * factd(j1 + m1) * factd(j2 - m2) * factd(j2 + m2)));
  double S = 0.0;
  for (int v = vmin; v <= vmax; ++v) {
    double t = factd(j2 + j3 + m1 - v) * factd(j1 - m1 + v)
             / (factd(v) * factd(j3 - j1 + j2 - v) * factd(j3 + m3 - v) * factd(v + j1 - j2 - m3));
    S += (((v + j2 + m2) & 1) ? -t : t);
  }
  return C * S;
}

// q_real_to_complex(l): complex (2l+1)x(2l+1)
__device__ void build_q(int l, double qr[9][9], double qi[9][9]) {
  for (int a = 0; a < 9; ++a)
    for (int b = 0; b < 9; ++b) { qr[a][b] = 0.0; qi[a][b] = 0.0; }
  const double s = 0.70710678118654752440;
  for (int m = -l; m < 0; ++m) {         // q[l+m][l+|m|]=s ; q[l+m][l-|m|]=-i*s
    qr[l + m][l - m] = s;
    qi[l + m][l + m] = -s;
  }
  qr[l][l] = 1.0;
  for (int m = 1; m <= l; ++m) {         // q[l+m][l+m]=(-1)^m s ; q[l+m][l-m]=i(-1)^m s
    double sg = (m & 1) ? -1.0 : 1.0;
    qr[l + m][l + m] = sg * s;
    qi[l + m][l - m] = sg * s;
  }
  // multiply by (-i)^l
  int ph = l & 3;
  double pr = (ph == 0) ? 1.0 : (ph == 2) ? -1.0 : 0.0;
  double pi = (ph == 1) ? -1.0 : (ph == 3) ? 1.0 : 0.0;
  int d = 2 * l + 1;
  for (int a = 0; a < d; ++a)
    for (int b = 0; b < d; ++b) {
      double r = qr[a][b], im = qi[a][b];
      qr[a][b] = r * pr - im * pi;
      qi[a][b] = r * pi + im * pr;
    }
}

// out[a][b][m] = (wigner_3j(l,l,l3) * sqrt(2*l3+1)) in the real basis
__device__ void wigner_block(int l, int l3, double out[5][5][9]) {
  const int d = 2 * l + 1, d3 = 2 * l3 + 1;
  double cg[5][5][9];
  for (int i = 0; i < 5; ++i) for (int k = 0; k < 5; ++k) for (int n = 0; n < 9; ++n) cg[i][k][n] = 0.0;
  for (int m1 = -l; m1 <= l; ++m1)
    for (int m2 = -l; m2 <= l; ++m2) {
      int m3 = m1 + m2;
      if (m3 >= -l3 && m3 <= l3)
        cg[l + m1][l + m2][l3 + m3] = su2_cg_coeff(l, m1, l, m2, l3, m3);
    }
  double q1r[9][9], q1i[9][9], q3r[9][9], q3i[9][9];
  build_q(l, q1r, q1i);
  build_q(l3, q3r, q3i);
  // C[j][lc][m] = Re( sum_{i,k,n} Q1[i][j] Q1[k][lc] conj(Q3[n][m]) cg[i][k][n] )
  double nrm = 0.0;
  for (int j = 0; j < d; ++j)
    for (int lc = 0; lc < d; ++lc)
      for (int m = 0; m < d3; ++m) {
        double ar = 0.0;
        for (int i = 0; i < d; ++i)
          for (int k = 0; k < d; ++k) {
            double t1r = q1r[i][j] * q1r[k][lc] - q1i[i][j] * q1i[k][lc];
            double t1i = q1r[i][j] * q1i[k][lc] + q1i[i][j] * q1r[k][lc];
            for (int n = 0; n < d3; ++n) {
              double c = cg[i][k][n];
              if (c == 0.0) continue;
              // Re(t1 * conj(Q3[n][m]))
              ar += c * (t1r * q3r[n][m] + t1i * q3i[n][m]);
            }
          }
        out[j][lc][m] = ar;
        nrm += ar * ar;
      }
  double scale = sqrt((double)d3) / sqrt(nrm);
  for (int j = 0; j < d; ++j)
    for (int lc = 0; lc < d; ++lc)
      for (int m = 0; m < d3; ++m) out[j][lc][m] *= scale;
}

// Fill padded WMMA B-matrix tables into workspace.
__global__ void init_cg_kernel(float* __restrict__ W) {
  const int tid = threadIdx.x;
  for (int i = tid; i < W_TOTAL; i += blockDim.x) W[i] = 0.0f;
  __syncthreads();
  if (tid == 0) {
    double t[5][5][9];
    const int koff1[3] = {0, 1, 4};
    for (int l3 = 0; l3 < 3; ++l3) {
      wigner_block(1, l3, t);
      for (int a = 0; a < 3; ++a)
        for (int b = 0; b < 3; ++b)
          for (int m = 0; m < 2 * l3 + 1; ++m)
            W[W1_OFF + (3 * a + b) * 16 + koff1[l3] + m] = (float)t[a][b][m];
    }
    const int koff2[5] = {0, 1, 4, 9, 16};
    for (int l3 = 0; l3 < 5; ++l3) {
      wigner_block(2, l3, t);
      for (int a = 0; a < 5; ++a)
        for (int b = 0; b < 5; ++b)
          for (int m = 0; m < 2 * l3 + 1; ++m)
            W[W2_OFF + (5 * a + b) * 32 + koff2[l3] + m] = (float)t[a][b][m];
    }
  }
}

// ===================== main bandwidth-bound ETP kernel =====================
// One node per 256-thread block (8 wave32). Stage inputs in LDS via b128 NT
// loads, form outer products, contract with CG via V_WMMA_F32_16X16X4_F32
// (one 16x16 tile per wave: 4 tiles for l=1 [64x12x16], 4 for l=2 [32x28x32]),
// scatter to an LDS row buffer, write the 1504-float row with b128 NT stores.

__global__ __launch_bounds__(256) void etp_wmma_kernel(
    const float* __restrict__ x1g, const float* __restrict__ x2g,
    const float* __restrict__ W, float* __restrict__ out) {
  __shared__ float sx1[DIM_IN], sx2[DIM_IN];
  __shared__ float o1[64 * 12];   // l=1 outer products, K padded 9->12
  __shared__ float o2[32 * 28];   // l=2 outer products, K padded 25->28
  __shared__ float obuf[DIM_OUT]; // staged output row

  const int node = blockIdx.x;
  const int tid = threadIdx.x;
  const long long ibase = (long long)node * DIM_IN;

  // ---- stage inputs: 120 float4 per row; waves 0-3 -> x1, waves 4-7 -> x2 ----
  {
    const v4f* x1v = (const v4f*)(x1g + ibase);
    const v4f* x2v = (const v4f*)(x2g + ibase);
    if (tid < 120) {
      *(v4f*)(sx1 + 4 * tid) = __builtin_nontemporal_load(x1v + tid);
    } else if (tid >= 128 && tid < 248) {
      int i = tid - 128;
      *(v4f*)(sx2 + 4 * i) = __builtin_nontemporal_load(x2v + i);
    }
  }
  __syncthreads();

  // l=0 chunk (output offset 0..127): pure elementwise
  if (tid < 128) obuf[tid] = sx1[tid] * sx2[tid];
  // outer products
  for (int idx = tid; idx < 64 * 12; idx += 256) {
    int u = idx / 12, p = idx % 12;
    o1[idx] = (p < 9) ? sx1[128 + 3 * u + p / 3] * sx2[128 + 3 * u + p % 3] : 0.0f;
  }
  for (int idx = tid; idx < 32 * 28; idx += 256) {
    int u = idx / 28, p = idx % 28;
    o2[idx] = (p < 25) ? sx1[320 + 5 * u + p / 5] * sx2[320 + 5 * u + p % 5] : 0.0f;
  }
  __syncthreads();

  const int wave = __builtin_amdgcn_readfirstlane((int)(tid >> 5)); // uniform -> scalar branch, EXEC stays all-1s
  const int lane = tid & 31;
  const int h = lane >> 4;   // half-wave (K split for A/B, M split for D)
  const int m = lane & 15;   // row (A) / col (B, D)
  v8f acc = {};

  if (wave < 4) {
    // l=1: rows u = 16*wave .. +15, cols k=0..8 (of 16), K = 12 -> 3 wmma steps
    const int rowbase = wave << 4;
#pragma unroll
    for (int s = 0; s < 3; ++s) {
      v2f A, B;
      A.x = o1[(rowbase + m) * 12 + 4 * s + 2 * h];
      A.y = o1[(rowbase + m) * 12 + 4 * s + 2 * h + 1];
      B.x = W[W1_OFF + (4 * s + 2 * h) * 16 + m];
      B.y = W[W1_OFF + (4 * s + 2 * h + 1) * 16 + m];
      acc = __builtin_amdgcn_wmma_f32_16x16x4_f32(false, A, false, B, (short)0, acc, false, false);
    }
    if (m < 9) {
      int base, d3, kk;
      if (m == 0)      { base = 128; d3 = 1; kk = 0; }     // 64x(l3=0)
      else if (m <= 3) { base = 224; d3 = 3; kk = m - 1; } // 64x(l3=1)
      else             { base = 512; d3 = 5; kk = m - 4; } // 64x(l3=2)
#pragma unroll
      for (int r = 0; r < 8; ++r) {
        int u = rowbase + 8 * h + r;          // D: VGPR r holds row r+8h, col m
        obuf[base + u * d3 + kk] = acc[r];
      }
    }
  } else {
    // l=2: 2x2 tiles over [32 rows x 25 cols], K = 28 -> 7 wmma steps
    const int v = wave - 4;
    const int rowbase = (v >> 1) << 4;
    const int colbase = (v & 1) << 4;
#pragma unroll
    for (int s = 0; s < 7; ++s) {
      v2f A, B;
      A.x = o2[(rowbase + m) * 28 + 4 * s + 2 * h];
      A.y = o2[(rowbase + m) * 28 + 4 * s + 2 * h + 1];
      B.x = W[W2_OFF + (4 * s + 2 * h) * 32 + colbase + m];
      B.y = W[W2_OFF + (4 * s + 2 * h + 1) * 32 + colbase + m];
      acc = __builtin_amdgcn_wmma_f32_16x16x4_f32(false, A, false, B, (short)0, acc, false, false);
    }
    const int k = colbase + m;
    if (k < 25) {
      int base, d3, kk;
      if (k == 0)       { base = 192;  d3 = 1; kk = 0; }
      else if (k <= 3)  { base = 416;  d3 = 3; kk = k - 1; }
      else if (k <= 8)  { base = 832;  d3 = 5; kk = k - 4; }
      else if (k <= 15) { base = 992;  d3 = 7; kk = k - 9; }
      else              { base = 1216; d3 = 9; kk = k - 16; }
#pragma unroll
      for (int r = 0; r < 8; ++r) {
        int u = rowbase + 8 * h + r;
        obuf[base + u * d3 + kk] = acc[r];
      }
    }
  }
  __syncthreads();

  // ---- coalesced b128 non-temporal writeback: 376 float4 per row ----
  {
    v4f* outv = (v4f*)(out + (long long)node * DIM_OUT);
#pragma unroll
    for (int i = tid; i < DIM_OUT / 4; i += 256) {
      v4f val = *(const v4f*)(obuf + 4 * i);
      __builtin_nontemporal_store(val, outv + i);
    }
  }
}

extern "C" void kernel_launch(void* const* d_in, const int* in_sizes, int n_in,
                              void* d_out, int out_size, void* d_ws, size_t ws_size,
                              hipStream_t stream) {
  const float* x1 = (const float*)d_in[0];
  const float* x2 = (const float*)d_in[1];
  float* out = (float*)d_out;
  float* W = (float*)d_ws;  // needs W_TOTAL*4 = 4352 bytes
  const int nnodes = in_sizes[0] / DIM_IN;

  init_cg_kernel<<<1, 128, 0, stream>>>(W);
  etp_wmma_kernel<<<nnodes, 256, 0, stream>>>(x1, x2, W, out);
}